// MACUnit_14259291423303
// MI455X (gfx1250) — compile-verified
//
#include <hip/hip_runtime.h>
#include <math.h>

typedef float v2f __attribute__((ext_vector_type(2)));
typedef float v8f __attribute__((ext_vector_type(8)));

#define DIM   512
#define BTOT  384      // B2*R = 64*6
#define KNUM  36
#define B2N   64
#define RN    6
#define NSTEP 12

#define WMMA_F32(a, b, c) \
    __builtin_amdgcn_wmma_f32_16x16x4_f32(false, (a), false, (b), (short)0, (c), false, false)

// ---------------------------------------------------------------------------
// WMMA f32 GEMM:  C[M,N] = act( [A0 | A1] @ W + bias )
//   A is logically M x Ktot; first K0 columns from A0 (ld lda0), the rest
//   from A1 (ld lda1).  W is Ktot x N row-major.  One wave computes a
//   32(M) x 64(N) register tile: 2 A-fragments x 4 B-fragments -> 8
//   V_WMMA_F32_16X16X4_F32 accumulators (K-step of 4).  All four B
//   fragments are preloaded per k-step so 10 loads are in flight before
//   the single wait, then 8 WMMAs issue back-to-back.
//   Fragment layouts per CDNA5 ISA 7.12.2:
//     A 16x4 : lanes 0-15 -> M=lane, v0=K, v1=K+1 ; lanes 16-31 -> K+2,K+3
//     B 4x16 : lanes 0-15 -> row K (v0), K+1 (v1); lanes 16-31 -> K+2, K+3
//     C 16x16: VGPR i -> M=i (lanes 0-15), M=i+8 (lanes 16-31), N=lane&15
//   Requires M % 32 == 0, N % 64 == 0, K0/Ktot % 4 == 0.
// act: 0 = none, 1 = tanh
// ---------------------------------------------------------------------------
__global__ void __launch_bounds__(256) gemm_wmma_f32(
    const float* __restrict__ A0, int lda0, int K0,
    const float* __restrict__ A1, int lda1,
    const float* __restrict__ W,
    const float* __restrict__ bias,
    float* __restrict__ C,
    int M, int N, int Ktot, int act)
{
    const int lane   = threadIdx.x & 31;
    const int wave   = threadIdx.x >> 5;
    const int tilesN = N >> 6;                 // 64-wide column tiles
    const int gw     = blockIdx.x * 8 + wave;
    if (gw >= (M >> 5) * tilesN) return;

    const int tm  = (gw / tilesN) << 5;        // 32-row tile origin
    const int tn  = (gw % tilesN) << 6;        // 64-col tile origin
    const int l16 = lane & 15;
    const int hi  = lane >> 4;                 // 0: lanes 0-15, 1: lanes 16-31
    const int row0 = tm + l16;
    const int row1 = row0 + 16;
    const int col  = tn + l16;

    v8f acc[2][4] = {};

    // W walker: lane reads rows (k+2*hi) and (k+2*hi+1) at col + 16*j
    const float* wp = W + (size_t)(hi << 1) * N + col;

    {   // --- first A source: K columns [0, K0) ---
        const float* ap0 = A0 + (size_t)row0 * lda0 + (hi << 1);
        const float* ap1 = A0 + (size_t)row1 * lda0 + (hi << 1);
        for (int k = 0; k < K0; k += 4) {
            v2f a0 = *(const v2f*)ap0; ap0 += 4;
            v2f a1 = *(const v2f*)ap1; ap1 += 4;
            v2f b[4];
#pragma unroll
            for (int j = 0; j < 4; ++j) {
                b[j].x = wp[j * 16];
                b[j].y = wp[j * 16 + N];
            }
#pragma unroll
            for (int j = 0; j < 4; ++j) {
                acc[0][j] = WMMA_F32(a0, b[j], acc[0][j]);
                acc[1][j] = WMMA_F32(a1, b[j], acc[1][j]);
            }
            wp += (size_t)4 * N;
        }
    }
    if (K0 < Ktot) {   // --- concat tail: K columns [K0, Ktot) from A1 ---
        const float* ap0 = A1 + (size_t)row0 * lda1 + (hi << 1);
        const float* ap1 = A1 + (size_t)row1 * lda1 + (hi << 1);
        for (int k = K0; k < Ktot; k += 4) {
            v2f a0 = *(const v2f*)ap0; ap0 += 4;
            v2f a1 = *(const v2f*)ap1; ap1 += 4;
            v2f b[4];
#pragma unroll
            for (int j = 0; j < 4; ++j) {
                b[j].x = wp[j * 16];
                b[j].y = wp[j * 16 + N];
            }
#pragma unroll
            for (int j = 0; j < 4; ++j) {
                acc[0][j] = WMMA_F32(a0, b[j], acc[0][j]);
                acc[1][j] = WMMA_F32(a1, b[j], acc[1][j]);
            }
            wp += (size_t)4 * N;
        }
    }

    // epilogue: bias + activation + store
#pragma unroll
    for (int t = 0; t < 2; ++t) {
        const int orow = tm + (t << 4) + (hi << 3);
#pragma unroll
        for (int j = 0; j < 4; ++j) {
            const int oc = col + j * 16;
            const float bv = bias ? bias[oc] : 0.0f;
#pragma unroll
            for (int i = 0; i < 8; ++i) {
                float v = acc[t][j][i] + bv;
                if (act == 1) v = tanhf(v);
                C[(size_t)(orow + i) * N + oc] = v;
            }
        }
    }
}

// ---------------------------------------------------------------------------
// final[b2,i,d] = sum_j mask[b2,i,j] * ctx[b2,j,d]      (tiny 6x6 mix)
// ---------------------------------------------------------------------------
__global__ void maskmix_kernel(const float* __restrict__ mask,  // (B2,R,R)
                               const float* __restrict__ ctx,   // (BTOT,DIM)
                               float* __restrict__ fin)         // (BTOT,DIM)
{
    const int idx  = blockIdx.x * 256 + threadIdx.x;   // < BTOT*DIM
    const int d    = idx & (DIM - 1);
    const int rowi = idx >> 9;                          // 0..BTOT-1
    const int b2   = rowi / RN;
    const int i    = rowi % RN;
    const float* mrow  = mask + ((size_t)b2 * RN + i) * RN;
    const float* cbase = ctx + (size_t)b2 * RN * DIM + d;
    float s = 0.0f;
#pragma unroll
    for (int j = 0; j < RN; ++j) s += mrow[j] * cbase[(size_t)j * DIM];
    fin[idx] = s;
}

// ---------------------------------------------------------------------------
// Fused attention + read:
//   s[b,k]   = sum_d relu(hk[b,k,d] + pqa[b,d]) * wa2[d] + ba2
//   att      = softmax_k(s)
//   read[b,:]= sum_k att[b,k] * knowledge[b,k,:]
// One block per b.
// ---------------------------------------------------------------------------
__global__ void __launch_bounds__(256) attn_read_kernel(
    const float* __restrict__ hk,        // (BTOT*KNUM, DIM)
    const float* __restrict__ pqa,       // (BTOT, DIM)
    const float* __restrict__ wa2,       // (DIM)
    const float* __restrict__ ba2p,      // (1)
    const float* __restrict__ knowledge, // (BTOT, KNUM, 4*DIM)
    float* __restrict__ readv)           // (BTOT, 4*DIM)
{
    __shared__ float s[KNUM];
    __shared__ float att[KNUM];
    const int b    = blockIdx.x;
    const int lane = threadIdx.x & 31;
    const int wave = threadIdx.x >> 5;
    const float* pq = pqa + (size_t)b * DIM;

    for (int k = wave; k < KNUM; k += 8) {
        const float* h = hk + ((size_t)b * KNUM + k) * DIM;
        float p = 0.0f;
        for (int d = lane; d < DIM; d += 32) {
            float t = h[d] + pq[d];
            t = t > 0.0f ? t : 0.0f;
            p += t * wa2[d];
        }
#pragma unroll
        for (int off = 16; off > 0; off >>= 1) p += __shfl_xor(p, off, 32);
        if (lane == 0) s[k] = p + ba2p[0];
    }
    __syncthreads();

    float m = -1e30f;
#pragma unroll
    for (int k = 0; k < KNUM; ++k) m = fmaxf(m, s[k]);
    float den = 0.0f;
#pragma unroll
    for (int k = 0; k < KNUM; ++k) den += expf(s[k] - m);
    if (threadIdx.x < KNUM) att[threadIdx.x] = expf(s[threadIdx.x] - m) / den;
    __syncthreads();

    const float* kb = knowledge + (size_t)b * KNUM * (4 * DIM);
    for (int c = threadIdx.x; c < 4 * DIM; c += 256) {
        float acc = 0.0f;
#pragma unroll 4
        for (int k = 0; k < KNUM; ++k)
            acc += att[k] * kb[(size_t)k * (4 * DIM) + c];
        readv[(size_t)b * (4 * DIM) + c] = acc;
    }
}

// mem[b,:] = mem0[0,:]
__global__ void bcast_mem_kernel(const float* __restrict__ src,
                                 float* __restrict__ dst)
{
    const int idx = blockIdx.x * 256 + threadIdx.x;   // < BTOT*DIM
    dst[idx] = src[idx & (DIM - 1)];
}

// ---------------------------------------------------------------------------
extern "C" void kernel_launch(void* const* d_in, const int* in_sizes, int n_in,
                              void* d_out, int out_size, void* d_ws, size_t ws_size,
                              hipStream_t stream)
{
    const float* question  = (const float*)d_in[0];   // (384, 512)
    const float* knowledge = (const float*)d_in[1];   // (384, 36, 2048)
    const float* cmask     = (const float*)d_in[2];   // (64, 6, 6, 1)
    // d_in[3] = adj (unused by reference)
    const float* mem0      = (const float*)d_in[4];   // (1, 512)
    const float* Wm = (const float*)d_in[5],  *bm  = (const float*)d_in[6];
    const float* Wq = (const float*)d_in[7],  *bq  = (const float*)d_in[8];
    const float* Wa1= (const float*)d_in[9],  *ba1 = (const float*)d_in[10];
    const float* Wa2= (const float*)d_in[11], *ba2 = (const float*)d_in[12];
    const float* Wr = (const float*)d_in[13], *br  = (const float*)d_in[14];
    const float* Wc = (const float*)d_in[15], *bc  = (const float*)d_in[16];
    float* out = (float*)d_out;

    // workspace carve-up (floats)
    float* ws = (float*)d_ws;
    size_t off = 0;
    float* hk      = ws + off; off += (size_t)BTOT * KNUM * DIM;  // 13824x512
    float* ctx     = ws + off; off += (size_t)BTOT * DIM;
    float* fin     = ws + off; off += (size_t)BTOT * DIM;
    float* projq   = ws + off; off += (size_t)BTOT * DIM;
    float* pqa     = ws + off; off += (size_t)BTOT * DIM;
    float* readv   = ws + off; off += (size_t)BTOT * 4 * DIM;
    float* readlow = ws + off; off += (size_t)BTOT * DIM;
    float* memA    = ws + off; off += (size_t)BTOT * DIM;
    float* memB    = ws + off; off += (size_t)BTOT * DIM;
    (void)ws_size; (void)in_sizes; (void)n_in; (void)out_size;

    auto gemm = [&](const float* A0, int lda0, int K0,
                    const float* A1, int lda1,
                    const float* W, const float* bias,
                    float* C, int M, int N, int Ktot, int act) {
        int tiles  = (M / 32) * (N / 64);
        int blocks = (tiles + 7) / 8;
        gemm_wmma_f32<<<blocks, 256, 0, stream>>>(A0, lda0, K0, A1, lda1,
                                                  W, bias, C, M, N, Ktot, act);
    };

    // memory0 = broadcast(mem0)
    bcast_mem_kernel<<<(BTOT * DIM) / 256, 256, 0, stream>>>(mem0, memA);

    // Precompute (step-invariant): hk = knowledge @ Wa1[:2048] + ba1
    gemm(knowledge, 4 * DIM, 4 * DIM, nullptr, 0,
         Wa1, ba1, hk, BTOT * KNUM, DIM, 4 * DIM, /*act=*/0);

    const float* Wa1q = Wa1 + (size_t)(4 * DIM) * DIM;  // rows 2048..2559

    for (int step = 0; step < NSTEP; ++step) {
        float* mcur  = (step & 1) ? memB : memA;
        float* mnext = (step == NSTEP - 1) ? out : ((step & 1) ? memA : memB);

        // ctx = tanh([question | mem] @ Wm + bm)
        gemm(question, DIM, DIM, mcur, DIM, Wm, bm, ctx, BTOT, DIM, 2 * DIM, 1);

        // final = mask-mix of ctx
        maskmix_kernel<<<(BTOT * DIM) / 256, 256, 0, stream>>>(cmask, ctx, fin);

        // projq = [final | question] @ Wq + bq
        gemm(fin, DIM, DIM, question, DIM, Wq, bq, projq, BTOT, DIM, 2 * DIM, 0);

        // pqa = projq @ Wa1[2048:]   (bias already folded into hk)
        gemm(projq, DIM, DIM, nullptr, 0, Wa1q, nullptr, pqa, BTOT, DIM, DIM, 0);

        // attention + weighted read of knowledge
        attn_read_kernel<<<BTOT, 256, 0, stream>>>(hk, pqa, Wa2, ba2,
                                                   knowledge, readv);

        // read_low = read @ Wr + br
        gemm(readv, 4 * DIM, 4 * DIM, nullptr, 0, Wr, br, readlow,
             BTOT, DIM, 4 * DIM, 0);

        // mem_next = [read_low | mem] @ Wc + bc
        gemm(readlow, DIM, DIM, mcur, DIM, Wc, bc, mnext, BTOT, DIM, 2 * DIM, 0);
    }
}